// RGCN_3186865733925
// MI455X (gfx1250) — compile-verified
//
#include <hip/hip_runtime.h>
#include <hip/hip_bf16.h>

typedef __attribute__((ext_vector_type(2))) float v2f;
typedef __attribute__((ext_vector_type(8))) float v8f;

#define NN 100000   // nodes
#define NE 300000   // edges per relation
#define NR 3        // relations
#define KDIM 128    // input feature dim (both layers)

// ---------------------------------------------------------------------------
// Degree accumulation: sOut[r][src]++ , sIn[r][dst]++  (float counts, exact)
// ---------------------------------------------------------------------------
__global__ void deg_kernel(const int* __restrict__ edges,
                           float* __restrict__ sOut, float* __restrict__ sIn) {
    int t = blockIdx.x * 256 + threadIdx.x;
    if (t >= NR * NE) return;
    int r = t / NE;
    int e = t - r * NE;
    int src = edges[(r * 2 + 0) * NE + e];
    int dst = edges[(r * 2 + 1) * NE + e];
    __hip_atomic_fetch_add(&sOut[r * NN + src], 1.0f, __ATOMIC_RELAXED, __HIP_MEMORY_SCOPE_AGENT);
    __hip_atomic_fetch_add(&sIn [r * NN + dst], 1.0f, __ATOMIC_RELAXED, __HIP_MEMORY_SCOPE_AGENT);
}

// deg -> clamp(deg,1)^{-1/2}, in place over both sOut and sIn (contiguous)
__global__ void scale_kernel(float* __restrict__ buf, int n) {
    int t = blockIdx.x * 256 + threadIdx.x;
    if (t >= n) return;
    buf[t] = rsqrtf(fmaxf(buf[t], 1.0f));
}

// ---------------------------------------------------------------------------
// Dense GEMM: M = diag(sOut) * X * W,  X:[NN x 128], W:[128 x F], M:[NN x F]
// Wave computes a 16x32 strip via two v_wmma_f32_16x16x4_f32 accumulators.
// Block: 256 threads = 8 waves -> 128 rows; grid.y picks a 32-column band.
// ---------------------------------------------------------------------------
template<int F>
__global__ __launch_bounds__(256)
void gemm_kernel(const float* __restrict__ X, const float* __restrict__ W,
                 const float* __restrict__ sOut, float* __restrict__ M) {
    // W tile [128 x 32] in LDS, XOR-16 column swizzle on (k>>1) parity so the
    // two 16-lane groups (K rows k and k+2) read disjoint bank halves.
    __shared__ float wt[KDIM * 32];
    const int tid  = threadIdx.x;
    const int nblk = blockIdx.y * 32;

    #pragma unroll
    for (int i = 0; i < 16; ++i) {
        int lin = i * 256 + tid;
        int k  = lin >> 5;
        int n2 = lin & 31;
        wt[k * 32 + (n2 ^ (((k >> 1) & 1) << 4))] = W[k * F + nblk + n2];
    }
    __syncthreads();

    const int wave  = tid >> 5;
    const int lane  = tid & 31;
    const int mbase = blockIdx.x * 128 + wave * 16;
    if (mbase >= NN) return;                 // wave-uniform: EXEC stays all-ones

    // f32 WMMA A layout: lane<16 -> row=lane, K={k0,k0+1}; lane>=16 -> K={k0+2,k0+3}
    const int   row  = mbase + (lane & 15);
    const int   koff = (lane >> 4) << 1;     // 0 or 2
    const int   ncol = lane & 15;
    const float s    = sOut[row];            // D_out^{-1/2}, constant over K loop
    const float* xr  = X + (size_t)row * KDIM;

    v8f acc0 = {};
    v8f acc1 = {};

    #pragma unroll
    for (int k0 = 0; k0 < KDIM; k0 += 4) {
        const int kk = k0 + koff;
        v2f a = *(const v2f*)(xr + kk);      // 8B aligned (kk even)
        a.x *= s; a.y *= s;

        const int sw0 = (((kk    ) >> 1) & 1) << 4;
        const int sw1 = (((kk + 1) >> 1) & 1) << 4;
        v2f b0, b1;
        b0.x = wt[(kk    ) * 32 + ( ncol       ^ sw0)];
        b0.y = wt[(kk + 1) * 32 + ( ncol       ^ sw1)];
        b1.x = wt[(kk    ) * 32 + ((ncol + 16) ^ sw0)];
        b1.y = wt[(kk + 1) * 32 + ((ncol + 16) ^ sw1)];

        acc0 = __builtin_amdgcn_wmma_f32_16x16x4_f32(false, a, false, b0,
                                                     (short)0, acc0, false, false);
        acc1 = __builtin_amdgcn_wmma_f32_16x16x4_f32(false, a, false, b1,
                                                     (short)0, acc1, false, false);
    }

    // D layout: vgpr v holds (m = v + 8*(lane>>4), n = lane&15)
    const int srow = mbase + ((lane >> 4) << 3);
    float* mp = M + (size_t)srow * F + nblk + ncol;
    #pragma unroll
    for (int v = 0; v < 8; ++v) {
        mp[(size_t)v * F]      = acc0[v];
        mp[(size_t)v * F + 16] = acc1[v];
    }
}

// ---------------------------------------------------------------------------
// Scaled scatter: out[dst] += sIn[dst] * M[src]   (per edge, F floats)
// One thread handles 4 contiguous floats (float4 gather + 4 f32 atomics).
// ---------------------------------------------------------------------------
template<int F>
__global__ void scatter_kernel(const float* __restrict__ M,
                               const int* __restrict__ src,
                               const int* __restrict__ dst,
                               const float* __restrict__ sIn,
                               float* __restrict__ out) {
    constexpr int CH = F >> 2;
    int t = blockIdx.x * 256 + threadIdx.x;
    int e = t / CH;
    int c = t - e * CH;
    if (e >= NE) return;
    int s = src[e];
    int d = dst[e];
    float sc = sIn[d];
    float4 v = *(const float4*)(M + (size_t)s * F + c * 4);
    float* o = out + (size_t)d * F + c * 4;
    __hip_atomic_fetch_add(o + 0, sc * v.x, __ATOMIC_RELAXED, __HIP_MEMORY_SCOPE_AGENT);
    __hip_atomic_fetch_add(o + 1, sc * v.y, __ATOMIC_RELAXED, __HIP_MEMORY_SCOPE_AGENT);
    __hip_atomic_fetch_add(o + 2, sc * v.z, __ATOMIC_RELAXED, __HIP_MEMORY_SCOPE_AGENT);
    __hip_atomic_fetch_add(o + 3, sc * v.w, __ATOMIC_RELAXED, __HIP_MEMORY_SCOPE_AGENT);
}

// ---------------------------------------------------------------------------
// h = relu(h + sum_r b[r][col])
// ---------------------------------------------------------------------------
template<int F>
__global__ void bias_relu_kernel(float* __restrict__ h, const float* __restrict__ b) {
    size_t t = (size_t)blockIdx.x * 256 + threadIdx.x;
    if (t >= (size_t)NN * F) return;
    int col = (int)(t & (F - 1));
    float bias = b[col] + b[F + col] + b[2 * F + col];
    h[t] = fmaxf(h[t] + bias, 0.0f);
}

// ---------------------------------------------------------------------------
extern "C" void kernel_launch(void* const* d_in, const int* in_sizes, int n_in,
                              void* d_out, int out_size, void* d_ws, size_t ws_size,
                              hipStream_t stream) {
    const float* x     = (const float*)d_in[0];   // [NN,128]
    const float* W1    = (const float*)d_in[1];   // [3,128,128]
    const float* b1    = (const float*)d_in[2];   // [3,128]
    const float* W2    = (const float*)d_in[3];   // [3,128,64]
    const float* b2    = (const float*)d_in[4];   // [3,64]
    const int*   edges = (const int*)d_in[5];     // [3,2,NE]
    float*       out   = (float*)d_out;           // [NN,64]

    float* ws   = (float*)d_ws;
    float* sOut = ws;                             // [3*NN] counts -> scales
    float* sIn  = ws + 3 * NN;                    // [3*NN]
    float* h    = ws + 6 * NN;                    // [NN,128]
    float* m    = h + (size_t)NN * KDIM;          // [NN,128] GEMM scratch

    (void)in_sizes; (void)n_in; (void)out_size; (void)ws_size;

    // degrees -> inverse-sqrt scales
    hipMemsetAsync(sOut, 0, (size_t)6 * NN * sizeof(float), stream);
    deg_kernel<<<(NR * NE + 255) / 256, 256, 0, stream>>>(edges, sOut, sIn);
    scale_kernel<<<(6 * NN + 255) / 256, 256, 0, stream>>>(sOut, 6 * NN);

    // ----- layer 1: F = 128 -----
    hipMemsetAsync(h, 0, (size_t)NN * 128 * sizeof(float), stream);
    dim3 g1((NN + 127) / 128, 128 / 32);
    for (int r = 0; r < NR; ++r) {
        gemm_kernel<128><<<g1, 256, 0, stream>>>(x, W1 + (size_t)r * 128 * 128,
                                                 sOut + r * NN, m);
        scatter_kernel<128><<<((size_t)NE * 32 + 255) / 256, 256, 0, stream>>>(
            m, edges + (r * 2) * NE, edges + (r * 2 + 1) * NE, sIn + r * NN, h);
    }
    bias_relu_kernel<128><<<(unsigned)(((size_t)NN * 128 + 255) / 256), 256, 0, stream>>>(h, b1);

    // ----- layer 2: F = 64 -----
    hipMemsetAsync(out, 0, (size_t)NN * 64 * sizeof(float), stream);
    dim3 g2((NN + 127) / 128, 64 / 32);
    for (int r = 0; r < NR; ++r) {
        gemm_kernel<64><<<g2, 256, 0, stream>>>(h, W2 + (size_t)r * 128 * 64,
                                                sOut + r * NN, m);
        scatter_kernel<64><<<((size_t)NE * 16 + 255) / 256, 256, 0, stream>>>(
            m, edges + (r * 2) * NE, edges + (r * 2 + 1) * NE, sIn + r * NN, out);
    }
    bias_relu_kernel<64><<<(unsigned)(((size_t)NN * 64 + 255) / 256), 256, 0, stream>>>(out, b2);
}